// AssignmentLoss_65867618452265
// MI455X (gfx1250) — compile-verified
//
#include <hip/hip_runtime.h>
#include <hip/hip_bf16.h>

// Problem constants (from reference)
constexpr int   BB       = 64;
constexpr int   NN       = 1024;
constexpr int   CC       = 558;
constexpr int   C1       = 559;    // with dustbin
constexpr int   ITERS    = 20;
constexpr float NEG_INF  = -1e30f;
constexpr float EPSV     = 1e-8f;
constexpr float CLASS_W  = 1.0f;
constexpr float ASSIGN_W = 0.5f;

typedef __attribute__((ext_vector_type(2))) float v2f;
typedef __attribute__((ext_vector_type(8))) float v8f;

__device__ __forceinline__ float wave_max32(float v) {
#pragma unroll
  for (int o = 16; o > 0; o >>= 1) v = fmaxf(v, __shfl_xor(v, o, 32));
  return v;
}
__device__ __forceinline__ float wave_sum32(float v) {
#pragma unroll
  for (int o = 16; o > 0; o >>= 1) v += __shfl_xor(v, o, 32);
  return v;
}

// ---------------------------------------------------------------------------
// k0: per-batch mask scan (one wave per b): nvis[b], tgt[b,n] = labels[b, rank],
//     and zero-init log_v (Sinkhorn starts with v = 0).
// ---------------------------------------------------------------------------
__global__ void k0_scan(const int* __restrict__ labels,
                        const unsigned char* __restrict__ mask,
                        int* __restrict__ tgt, float* __restrict__ nvis,
                        float* __restrict__ log_v) {
  const int b = blockIdx.x;
  const int lane = threadIdx.x;
  for (int c = lane; c < C1; c += 32) log_v[b * C1 + c] = 0.0f;
  int base = 0;
  for (int chunk = 0; chunk < NN / 32; ++chunk) {
    const int n = chunk * 32 + lane;
    const int m = mask[b * NN + n] ? 1 : 0;
    const unsigned int b32 = (unsigned int)__ballot(m);  // wave32: low 32 bits
    const unsigned int le = (lane == 31) ? 0xFFFFFFFFu : ((1u << (lane + 1)) - 1u);
    const int cum = base + __popc(b32 & le);             // inclusive cumsum
    const int rank = (cum > 0) ? (cum - 1) : 0;          // clip(cumsum-1, 0)
    tgt[b * NN + n] = labels[b * NN + rank];
    base += __popc(b32);
  }
  if (lane == 0) nvis[b] = (float)base;
}

// ---------------------------------------------------------------------------
// k1: one wave per (b,n) row: log-softmax over C, write ls (+dustbin),
//     log_mu, and masked NLL term. ls lands in L2 (146.5MB < 192MB).
// ---------------------------------------------------------------------------
__global__ void k1_logsoftmax(const float* __restrict__ logits,
                              const float* __restrict__ dust,
                              const unsigned char* __restrict__ mask,
                              const int* __restrict__ tgt,
                              const float* __restrict__ nvis,
                              float* __restrict__ ls, float* __restrict__ log_mu,
                              float* __restrict__ nllm) {
  const int warp = threadIdx.x >> 5, lane = threadIdx.x & 31;
  const int row = blockIdx.x * 8 + warp;     // row = b*N + n
  const int b = row >> 10;
  const float* lg = logits + (size_t)row * CC;
  float mx = -3.4e38f;
  for (int c = lane; c < CC; c += 32) mx = fmaxf(mx, lg[c]);
  mx = wave_max32(mx);
  float s = 0.0f;
  for (int c = lane; c < CC; c += 32) s += __expf(lg[c] - mx);
  s = wave_sum32(s);
  const float logZ = mx + __logf(s);
  float* lsr = ls + (size_t)row * C1;
  for (int c = lane; c < CC; c += 32) lsr[c] = lg[c] - logZ;  // TEMP == 1
  if (lane == 0) {
    lsr[CC] = dust[0];                                        // dustbin / TEMP
    const int vis = mask[row] ? 1 : 0;
    log_mu[row] = vis ? -__logf(nvis[b]) : NEG_INF;
    nllm[row] = vis ? (logZ - lg[tgt[row]]) : 0.0f;           // -log_prob[tgt]
  }
}

// ---------------------------------------------------------------------------
// k2: Sinkhorn u-update. One wave handles 16 rows; the sum-exp over the 559
//     columns is accumulated on the matrix pipe via V_WMMA_F32_16X16X4_F32
//     with an all-ones B matrix (full f32 precision), co-executing with the
//     v_exp_f32 stream on the trans pipe.
//     A layout (16x4 f32): lanes 0-15 -> M=lane, K=0,1 ; lanes 16-31 -> K=2,3.
//     D row m (replicated across its 16 lanes) = sum_c exp(ls + v).
// ---------------------------------------------------------------------------
__global__ void k2_urow_wmma(const float* __restrict__ ls,
                             const float* __restrict__ log_v,
                             const float* __restrict__ log_mu,
                             float* __restrict__ log_u) {
  __shared__ float vsh[C1];
  const int b = blockIdx.x >> 3;                    // 8 blocks per batch
  for (int i = threadIdx.x; i < C1; i += 256) vsh[i] = log_v[b * C1 + i];
  __syncthreads();
  const int warp = threadIdx.x >> 5, lane = threadIdx.x & 31;
  const int n0 = ((blockIdx.x & 7) * 8 + warp) * 16;  // 16 rows per wave
  const int m = lane & 15;
  const int kb = (lane >> 4) * 2;
  const float* lsr = ls + ((size_t)(b * NN + n0 + m)) * C1;
  v2f bones; bones.x = 1.0f; bones.y = 1.0f;        // B = ones(4x16)
  v8f acc = {};
  for (int c0 = 0; c0 < C1; c0 += 4) {
    const int ca = c0 + kb, cb = ca + 1;
    v2f a;
    a.x = (ca < C1) ? __expf(lsr[ca] + vsh[ca]) : 0.0f;
    a.y = (cb < C1) ? __expf(lsr[cb] + vsh[cb]) : 0.0f;
    acc = __builtin_amdgcn_wmma_f32_16x16x4_f32(
        /*neg_a=*/false, a, /*neg_b=*/false, bones,
        /*c_mod=*/(short)0, acc, /*reuse_a=*/false, /*reuse_b=*/false);
  }
  // D layout: VGPR j @ lane 0 -> row j ; VGPR j @ lane 16 -> row 8+j.
  const int rbase = b * NN + n0;
  if (lane == 0) {
#pragma unroll
    for (int j = 0; j < 8; ++j) {
      const int r = rbase + j;
      log_u[r] = log_mu[r] - __logf(acc[j]);
    }
  }
  if (lane == 16) {
#pragma unroll
    for (int j = 0; j < 8; ++j) {
      const int r = rbase + 8 + j;
      log_u[r] = log_mu[r] - __logf(acc[j]);
    }
  }
}

// ---------------------------------------------------------------------------
// k3: Sinkhorn v-update. One thread per column c; coalesced 256-wide reads of
//     ls across c, log_u staged in LDS. Streams ls once from L2.
// ---------------------------------------------------------------------------
__global__ void k3_vcol(const float* __restrict__ ls,
                        const float* __restrict__ log_u,
                        float* __restrict__ log_v) {
  __shared__ float ush[NN];
  const int b = blockIdx.x / 3, cblk = blockIdx.x % 3;
  for (int i = threadIdx.x; i < NN; i += 256) ush[i] = log_u[b * NN + i];
  __syncthreads();
  const int c = cblk * 256 + (int)threadIdx.x;
  if (c >= C1) return;
  const float* lsb = ls + (size_t)b * NN * C1;
  float s = 0.0f;
  for (int n = 0; n < NN; ++n) s += __expf(lsb[(size_t)n * C1 + c] + ush[n]);
  log_v[b * C1 + c] = -__logf((float)C1) - __logf(s);   // log_nu - lse
}

// ---------------------------------------------------------------------------
// k4: entropy of soft assignment (dustbin column excluded), masked per row.
// ---------------------------------------------------------------------------
__global__ void k4_entropy(const float* __restrict__ ls,
                           const float* __restrict__ log_u,
                           const float* __restrict__ log_v,
                           const unsigned char* __restrict__ mask,
                           float* __restrict__ entm) {
  __shared__ float vsh[C1];
  const int warp = threadIdx.x >> 5, lane = threadIdx.x & 31;
  const int row = blockIdx.x * 8 + warp;
  const int b = row >> 10;
  for (int i = threadIdx.x; i < C1; i += 256) vsh[i] = log_v[b * C1 + i];
  __syncthreads();
  const float* lsr = ls + (size_t)row * C1;
  const float u = log_u[row];
  float e = 0.0f;
  for (int c = lane; c < CC; c += 32) {         // soft = P[:, :, :-1]
    const float p = __expf(lsr[c] + u + vsh[c]);
    e += p * __logf(p + EPSV);
  }
  e = wave_sum32(e);
  if (lane == 0) entm[row] = mask[row] ? -e : 0.0f;
}

// ---------------------------------------------------------------------------
// k5: deterministic final reduction (no float atomics): per-b sums of masked
//     nll & entropy, divide by nvis, mean over b, weighted combine.
// ---------------------------------------------------------------------------
__global__ void k5_final(const float* __restrict__ nllm,
                         const float* __restrict__ entm,
                         const float* __restrict__ nvis,
                         float* __restrict__ out) {
  __shared__ float r1[256], r2[256];
  float cls = 0.0f, asg = 0.0f;
  for (int b = 0; b < BB; ++b) {
    float s1 = 0.0f, s2 = 0.0f;
    for (int n = threadIdx.x; n < NN; n += 256) {
      s1 += nllm[b * NN + n];
      s2 += entm[b * NN + n];
    }
    r1[threadIdx.x] = s1; r2[threadIdx.x] = s2;
    __syncthreads();
    for (int o = 128; o > 0; o >>= 1) {
      if ((int)threadIdx.x < o) {
        r1[threadIdx.x] += r1[threadIdx.x + o];
        r2[threadIdx.x] += r2[threadIdx.x + o];
      }
      __syncthreads();
    }
    if (threadIdx.x == 0) { cls += r1[0] / nvis[b]; asg += r2[0] / nvis[b]; }
    __syncthreads();
  }
  if (threadIdx.x == 0)
    out[0] = (CLASS_W * cls + ASSIGN_W * asg) / (float)BB;
}

// ---------------------------------------------------------------------------
extern "C" void kernel_launch(void* const* d_in, const int* in_sizes, int n_in,
                              void* d_out, int out_size, void* d_ws, size_t ws_size,
                              hipStream_t stream) {
  (void)in_sizes; (void)n_in; (void)out_size; (void)ws_size;
  const float*         logits = (const float*)d_in[0];
  const float*         dust   = (const float*)d_in[1];
  const int*           labels = (const int*)d_in[2];
  const unsigned char* mask   = (const unsigned char*)d_in[3];
  float* out = (float*)d_out;

  float* w = (float*)d_ws;
  size_t o = 0;
  float* ls     = w + o; o += (size_t)BB * NN * C1;   // 146.5 MB -> L2 resident
  float* log_u  = w + o; o += (size_t)BB * NN;
  float* log_mu = w + o; o += (size_t)BB * NN;
  float* log_v  = w + o; o += (size_t)BB * C1;
  float* nllm   = w + o; o += (size_t)BB * NN;
  float* entm   = w + o; o += (size_t)BB * NN;
  float* nvis   = w + o; o += (size_t)BB;
  int*   tgt    = (int*)(w + o); o += (size_t)BB * NN;

  k0_scan<<<BB, 32, 0, stream>>>(labels, mask, tgt, nvis, log_v);
  k1_logsoftmax<<<BB * NN / 8, 256, 0, stream>>>(logits, dust, mask, tgt, nvis,
                                                 ls, log_mu, nllm);
  for (int it = 0; it < ITERS; ++it) {
    k2_urow_wmma<<<BB * 8, 256, 0, stream>>>(ls, log_v, log_mu, log_u);
    k3_vcol<<<BB * 3, 256, 0, stream>>>(ls, log_u, log_v);
  }
  k4_entropy<<<BB * NN / 8, 256, 0, stream>>>(ls, log_u, log_v, mask, entm);
  k5_final<<<1, 256, 0, stream>>>(nllm, entm, nvis, out);
}